// HeteroMoEFeedForward_23553600651637
// MI455X (gfx1250) — compile-verified
//
#include <hip/hip_runtime.h>
#include <math.h>

typedef __attribute__((ext_vector_type(2))) float v2f;
typedef __attribute__((ext_vector_type(8))) float v8f;

#define D_MODEL     1024
#define NUM_EXPERTS 8
#define TOKENS      8192
#define LN_EPS      1e-5f

// LDS tile geometry (per buffer)
#define A_ROWS   128
#define A_STRIDE 18          // 16 K-words + 2 pad -> conflict-free b64 reads
#define A_WORDS  (A_ROWS * A_STRIDE)     // 2304 words
#define B_WORDS  (16 * 64)               // 1024 words

// ---------------------------------------------------------------------------
// CDNA5 helpers
// ---------------------------------------------------------------------------
__device__ __forceinline__ v8f wmma_f32(v8f c, v2f a, v2f b) {
    // (neg_a, A, neg_b, B, c_mod, C, reuse_a, reuse_b)
    return __builtin_amdgcn_wmma_f32_16x16x4_f32(false, a, false, b,
                                                 (short)0, c, false, false);
}

// Low 32 bits of a generic pointer to __shared__ == wave-relative LDS byte addr
__device__ __forceinline__ unsigned lds_u32(const void* p) {
    return (unsigned)(unsigned long long)p;
}

// Async copy: per-lane 16B global -> LDS, tracked with ASYNCcnt (gfx1250)
__device__ __forceinline__ void async_b128(const float* gaddr, unsigned lds_byte) {
    asm volatile("global_load_async_to_lds_b128 %0, %1, off"
                 :: "v"(lds_byte), "v"(gaddr)
                 : "memory");
}

// ---------------------------------------------------------------------------
// Router: logits = x @ router_w + router_b; softmax; top-2 -> dense gates[T,8]
// ---------------------------------------------------------------------------
__global__ void __launch_bounds__(256)
router_kernel(const float* __restrict__ x,       // [T, D]
              const float* __restrict__ rw,      // [D, E]
              const float* __restrict__ rb,      // [E]
              float* __restrict__ gates) {       // [T, E]
    const int wave = threadIdx.x >> 5;
    const int lane = threadIdx.x & 31;
    const int t = blockIdx.x * 8 + wave;
    const float* xr = x + (size_t)t * D_MODEL;

    float acc[NUM_EXPERTS];
#pragma unroll
    for (int e = 0; e < NUM_EXPERTS; ++e) acc[e] = 0.f;

    for (int d = lane; d < D_MODEL; d += 32) {
        const float xv = xr[d];
        const float* wr = rw + (size_t)d * NUM_EXPERTS;
#pragma unroll
        for (int e = 0; e < NUM_EXPERTS; ++e) acc[e] += xv * wr[e];
    }
#pragma unroll
    for (int e = 0; e < NUM_EXPERTS; ++e) {
#pragma unroll
        for (int off = 16; off > 0; off >>= 1)
            acc[e] += __shfl_xor(acc[e], off, 32);
    }

    if (lane == 0) {
        float p[NUM_EXPERTS];
        float mx = -INFINITY;
#pragma unroll
        for (int e = 0; e < NUM_EXPERTS; ++e) {
            p[e] = acc[e] + rb[e];
            mx = fmaxf(mx, p[e]);
        }
        float denom = 0.f;
#pragma unroll
        for (int e = 0; e < NUM_EXPERTS; ++e) { p[e] = __expf(p[e] - mx); denom += p[e]; }
        const float inv = 1.f / denom;
#pragma unroll
        for (int e = 0; e < NUM_EXPERTS; ++e) p[e] *= inv;

        int i1 = 0;
#pragma unroll
        for (int e = 1; e < NUM_EXPERTS; ++e) if (p[e] > p[i1]) i1 = e;
        int i2 = (i1 == 0) ? 1 : 0;
#pragma unroll
        for (int e = 0; e < NUM_EXPERTS; ++e)
            if (e != i1 && p[e] > p[i2]) i2 = e;

        float* gr = gates + (size_t)t * NUM_EXPERTS;
#pragma unroll
        for (int e = 0; e < NUM_EXPERTS; ++e)
            gr[e] = (e == i1) ? p[i1] : ((e == i2) ? p[i2] : 0.f);
    }
}

// ---------------------------------------------------------------------------
// GEMM core: block = 8 waves -> 128x64 tile; wave -> 32x32 (2x2 f32 WMMA).
// A[128x16] and B[16x64] chunks staged in LDS via async global->LDS copies,
// double-buffered (prefetch chunk c+1 overlaps WMMA on chunk c).
// ---------------------------------------------------------------------------
__device__ __forceinline__ void
moe_gemm_core(const float* __restrict__ Amat,  // [M, K] row-major
              const float* __restrict__ W,     // [K, ldw] row-major
              int K, int ldw, int mBase, int nBase,
              float* smA, float* smB, v8f acc[2][2]) {
    const int tid  = threadIdx.x;
    const int lane = tid & 31;
    const int wave = tid >> 5;
    const int wm = wave >> 1, wn = wave & 1;
    const int kh  = lane >> 4;          // half-wave: K-pair (A/B), M+8 (C)
    const int col = lane & 15;

    // async-copy work assignment (256 threads)
    const int rowA0 = tid >> 2, segA = tid & 3;   // A tile: 128 rows x 4 segs
    const int rowA1 = rowA0 + 64;                 // second half of A tile
    const int rowB  = tid >> 4, segB = tid & 15;  // B tile: 16 rows x 16 segs

    const float* gA0 = Amat + (size_t)(mBase + rowA0) * K + segA * 4;
    const float* gA1 = Amat + (size_t)(mBase + rowA1) * K + segA * 4;
    const float* gB  = W + (size_t)rowB * ldw + nBase + segB * 4;

    const unsigned lA0 = lds_u32(smA + rowA0 * A_STRIDE + segA * 4);
    const unsigned lA1 = lds_u32(smA + rowA1 * A_STRIDE + segA * 4);
    const unsigned lB  = lds_u32(smB + rowB * 64 + segB * 4);

    const int NC = K >> 4;              // 16-wide K chunks

    auto issue = [&](int c, int buf) {
        const size_t kofs = (size_t)c << 4;                  // c*16 elements
        const unsigned ofsA = buf ? (unsigned)(A_WORDS * 4) : 0u;
        const unsigned ofsB = buf ? (unsigned)(B_WORDS * 4) : 0u;
        async_b128(gA0 + kofs, lA0 + ofsA);
        async_b128(gA1 + kofs, lA1 + ofsA);
        async_b128(gB + kofs * ldw, lB + ofsB);
    };

    issue(0, 0);
    for (int c = 0; c < NC; ++c) {
        const int cur = c & 1;
        if (c + 1 < NC) {
            issue(c + 1, cur ^ 1);
            asm volatile("s_wait_asynccnt 0x3" ::: "memory");  // chunk c done
        } else {
            asm volatile("s_wait_asynccnt 0x0" ::: "memory");
        }
        __syncthreads();

        const float* Ab = smA + cur * A_WORDS;
        const float* Bb = smB + cur * B_WORDS;
        const float* ar0 = Ab + (wm * 32 + col) * A_STRIDE + 2 * kh;
        const float* ar1 = ar0 + 16 * A_STRIDE;
        const float* bb  = Bb + wn * 32 + col;
#pragma unroll
        for (int kk = 0; kk < 16; kk += 4) {
            v2f a0 = *(const v2f*)(ar0 + kk);
            v2f a1 = *(const v2f*)(ar1 + kk);
            const float* br = bb + (kk + 2 * kh) * 64;
            v2f bA, bB;
            bA[0] = br[0];  bA[1] = br[64];
            bB[0] = br[16]; bB[1] = br[80];
            acc[0][0] = wmma_f32(acc[0][0], a0, bA);
            acc[0][1] = wmma_f32(acc[0][1], a0, bB);
            acc[1][0] = wmma_f32(acc[1][0], a1, bA);
            acc[1][1] = wmma_f32(acc[1][1], a1, bB);
        }
        __syncthreads();   // protect buffer reuse two iterations later
    }
}

// ---------------------------------------------------------------------------
// GEMM1: H = gelu(X[T,K] @ W1[K,Hd] + b1)
// ---------------------------------------------------------------------------
__global__ void __launch_bounds__(256)
gemm1_gelu(const float* __restrict__ X, const float* __restrict__ W,
           const float* __restrict__ bias, float* __restrict__ Hout,
           int K, int Hd) {
    __shared__ float smA[2 * A_WORDS];
    __shared__ float smB[2 * B_WORDS];
    const int lane = threadIdx.x & 31;
    const int wave = threadIdx.x >> 5;
    const int wm = wave >> 1, wn = wave & 1;
    const int kh = lane >> 4, col = lane & 15;
    const int m0 = blockIdx.x * 128 + wm * 32;
    const int n0 = blockIdx.y * 64 + wn * 32;

    v8f acc[2][2] = {};
    moe_gemm_core(X, W, K, Hd, blockIdx.x * 128, blockIdx.y * 64, smA, smB, acc);

#pragma unroll
    for (int mi = 0; mi < 2; ++mi) {
#pragma unroll
        for (int ni = 0; ni < 2; ++ni) {
            const int n = n0 + ni * 16 + col;
            const float bv = bias[n];
#pragma unroll
            for (int j = 0; j < 8; ++j) {
                const int m = m0 + mi * 16 + j + 8 * kh;
                const float v = acc[mi][ni][j] + bv;
                Hout[(size_t)m * Hd + n] =
                    0.5f * v * (1.f + erff(v * 0.70710678118654752f));
            }
        }
    }
}

// ---------------------------------------------------------------------------
// GEMM2: out_acc (+)= gate[t,e] * (H[T,K] @ W2[K,D] + b2)
// ---------------------------------------------------------------------------
__global__ void __launch_bounds__(256)
gemm2_acc(const float* __restrict__ A, const float* __restrict__ W,
          const float* __restrict__ bias, const float* __restrict__ gates,
          float* __restrict__ out_acc, int K, int expert, int first) {
    __shared__ float smA[2 * A_WORDS];
    __shared__ float smB[2 * B_WORDS];
    const int N = D_MODEL;
    const int lane = threadIdx.x & 31;
    const int wave = threadIdx.x >> 5;
    const int wm = wave >> 1, wn = wave & 1;
    const int kh = lane >> 4, col = lane & 15;
    const int m0 = blockIdx.x * 128 + wm * 32;
    const int n0 = blockIdx.y * 64 + wn * 32;

    v8f acc[2][2] = {};
    moe_gemm_core(A, W, K, N, blockIdx.x * 128, blockIdx.y * 64, smA, smB, acc);

#pragma unroll
    for (int mi = 0; mi < 2; ++mi) {
#pragma unroll
        for (int ni = 0; ni < 2; ++ni) {
            const int n = n0 + ni * 16 + col;
            const float bv = bias[n];
#pragma unroll
            for (int j = 0; j < 8; ++j) {
                const int m = m0 + mi * 16 + j + 8 * kh;
                const float g = gates[(size_t)m * NUM_EXPERTS + expert];
                const float contrib = g * (acc[mi][ni][j] + bv);
                const size_t idx = (size_t)m * N + n;
                out_acc[idx] = first ? contrib : (out_acc[idx] + contrib);
            }
        }
    }
}

// ---------------------------------------------------------------------------
// Final: out = x + LayerNorm(out_acc) * gamma + beta.  One block per token.
// ---------------------------------------------------------------------------
__global__ void __launch_bounds__(256)
ln_residual(const float* __restrict__ x, const float* __restrict__ o,
            const float* __restrict__ gamma, const float* __restrict__ beta,
            float* __restrict__ out) {
    const int t = blockIdx.x;
    const int lane = threadIdx.x & 31;
    const int wave = threadIdx.x >> 5;
    __shared__ float s1[8], s2[8];

    const float* row = o + (size_t)t * D_MODEL;
    float s = 0.f, sq = 0.f;
    for (int d = threadIdx.x; d < D_MODEL; d += 256) {
        const float v = row[d];
        s += v; sq += v * v;
    }
#pragma unroll
    for (int off = 16; off > 0; off >>= 1) {
        s += __shfl_xor(s, off, 32);
        sq += __shfl_xor(sq, off, 32);
    }
    if (lane == 0) { s1[wave] = s; s2[wave] = sq; }
    __syncthreads();
    float ts = 0.f, tsq = 0.f;
#pragma unroll
    for (int w = 0; w < 8; ++w) { ts += s1[w]; tsq += s2[w]; }
    const float mu = ts * (1.f / D_MODEL);
    const float var = tsq * (1.f / D_MODEL) - mu * mu;
    const float inv = rsqrtf(var + LN_EPS);

    const float* xr = x + (size_t)t * D_MODEL;
    float* orow = out + (size_t)t * D_MODEL;
    for (int d = threadIdx.x; d < D_MODEL; d += 256)
        orow[d] = xr[d] + (row[d] - mu) * inv * gamma[d] + beta[d];
}

// ---------------------------------------------------------------------------
// Host launcher
// ---------------------------------------------------------------------------
extern "C" void kernel_launch(void* const* d_in, const int* in_sizes, int n_in,
                              void* d_out, int out_size, void* d_ws, size_t ws_size,
                              hipStream_t stream) {
    static const int HID[NUM_EXPERTS] = {4096, 2048, 4096, 1024, 4096, 4096, 4096, 4096};

    // Input order: x, router_w, router_b, w1[0..7], b1[0..7], w2[0..7], b2[0..7], gamma, beta
    const float* x        = (const float*)d_in[0];
    const float* router_w = (const float*)d_in[1];
    const float* router_b = (const float*)d_in[2];
    const float* w1[NUM_EXPERTS]; const float* b1[NUM_EXPERTS];
    const float* w2[NUM_EXPERTS]; const float* b2[NUM_EXPERTS];
    for (int e = 0; e < NUM_EXPERTS; ++e) {
        w1[e] = (const float*)d_in[3 + e];
        b1[e] = (const float*)d_in[3 + NUM_EXPERTS + e];
        w2[e] = (const float*)d_in[3 + 2 * NUM_EXPERTS + e];
        b2[e] = (const float*)d_in[3 + 3 * NUM_EXPERTS + e];
    }
    const float* ln_gamma = (const float*)d_in[3 + 4 * NUM_EXPERTS];
    const float* ln_beta  = (const float*)d_in[4 + 4 * NUM_EXPERTS];
    float* out = (float*)d_out;

    // Workspace carve-up
    float* gates   = (float*)d_ws;                           // T*8
    float* out_acc = gates + (size_t)TOKENS * NUM_EXPERTS;   // T*D
    float* Hbuf    = out_acc + (size_t)TOKENS * D_MODEL;     // T*4096 (max h)

    router_kernel<<<TOKENS / 8, 256, 0, stream>>>(x, router_w, router_b, gates);

    for (int e = 0; e < NUM_EXPERTS; ++e) {
        const int h = HID[e];
        dim3 g1(TOKENS / 128, h / 64);
        gemm1_gelu<<<g1, 256, 0, stream>>>(x, w1[e], b1[e], Hbuf, D_MODEL, h);
        dim3 g2(TOKENS / 128, D_MODEL / 64);
        gemm2_acc<<<g2, 256, 0, stream>>>(Hbuf, w2[e], b2[e], gates, out_acc,
                                          h, e, (e == 0) ? 1 : 0);
    }

    ln_residual<<<TOKENS, 256, 0, stream>>>(x, out_acc, ln_gamma, ln_beta, out);
}